// NodeFeatures_58274116272364
// MI455X (gfx1250) — compile-verified
//
#include <hip/hip_runtime.h>

typedef __attribute__((ext_vector_type(2))) float v2f;
typedef __attribute__((ext_vector_type(4))) float v4f;
typedef __attribute__((ext_vector_type(8))) float v8f;

// Problem constants from the reference
constexpr int Bb = 8, Nn = 256, Dd = 128;
constexpr int R  = Bb * Nn;            // 2048 flattened (b,n) rows
constexpr int MT = R  / 16;            // 128 M-tiles
constexpr int NT = Dd / 16;            // 8   N-tiles

// ---------------------------------------------------------------------------
// Kernel 1: fp32 WMMA projections.
//   Ux = x @ Wu + bu  -> out   (B*N x D)
//   Vx = x @ Wv + bv  -> ws    (B*N x D)
// One wave per 16x16 output tile; K=128 consumed in 32 steps of
// v_wmma_f32_16x16x4_f32. A-fragment (16x4 f32, ISA 7.12.2):
//   lanes 0-15 : M=lane,  VGPR0=K0, VGPR1=K0+1
//   lanes 16-31: M=lane-16,VGPR0=K0+2,VGPR1=K0+3   -> one float2 load per lane.
// B-fragment (4x16) mirrored: lane%16 = N column, lane/16 selects K-pair.
// ---------------------------------------------------------------------------
__global__ __launch_bounds__(128) void proj_wmma_kernel(
    const float* __restrict__ x,
    const float* __restrict__ Wu, const float* __restrict__ bu,
    const float* __restrict__ Wv, const float* __restrict__ bv,
    float* __restrict__ Ux, float* __restrict__ Vx)
{
    const int lane  = threadIdx.x & 31;
    const int wid   = threadIdx.x >> 5;
    const int tile  = blockIdx.x * 4 + wid;      // 0..1023, exact cover
    const int row0  = (tile >> 3) * 16;          // M-tile base row
    const int col0  = (tile & 7)  * 16;          // N-tile base col
    const int half  = lane >> 4;                 // 0: K+{0,1}, 1: K+{2,3}
    const int lm    = lane & 15;

    v8f accU = {};
    v8f accV = {};

    const float* xrow = x + (size_t)(row0 + lm) * Dd;

    #pragma unroll 4
    for (int k0 = 0; k0 < Dd; k0 += 4) {
        const int ka = k0 + 2 * half;
        // A fragment: two consecutive K values for this lane's row
        v2f a = *(const v2f*)(xrow + ka);
        // B fragments: W[k][col0+lm] for k = ka, ka+1
        v2f bu2, bv2;
        bu2.x = Wu[(size_t)ka       * Dd + col0 + lm];
        bu2.y = Wu[(size_t)(ka + 1) * Dd + col0 + lm];
        bv2.x = Wv[(size_t)ka       * Dd + col0 + lm];
        bv2.y = Wv[(size_t)(ka + 1) * Dd + col0 + lm];

        accU = __builtin_amdgcn_wmma_f32_16x16x4_f32(
            false, a, false, bu2, (short)0, accU, false, false);
        accV = __builtin_amdgcn_wmma_f32_16x16x4_f32(
            false, a, false, bv2, (short)0, accV, false, false);
    }

    // C/D layout: VGPR i -> M = row0 + i + 8*half, N = col0 + lm.
    // Bias depends only on N, i.e. constant per lane.
    const float biasU = bu[col0 + lm];
    const float biasV = bv[col0 + lm];

    #pragma unroll
    for (int i = 0; i < 8; ++i) {
        const size_t o = (size_t)(row0 + i + 8 * half) * Dd + col0 + lm;
        Ux[o] = accU[i] + biasU;
        Vx[o] = accV[i] + biasV;
    }
}

// ---------------------------------------------------------------------------
// Kernel 2: fused per-channel edge gating + mean aggregation.
//   out[b,i,d] = Ux[b,i,d] + (sum_j g[b,i,j,d]*Vx[b,j,d]) / (1e-20 + sum_j g)
// One block per (b,i). 256 threads = 8 j-groups x 32 lanes; each lane owns a
// float4 of channels. edge_gate (268 MB) streams once with non-temporal loads
// so the reused Vx (128 KB/batch) stays hot in the 192 MB L2. Single pass fuses
// both reductions. LDS combines the 8 group partials.
// ---------------------------------------------------------------------------
__global__ __launch_bounds__(256) void gate_agg_kernel(
    const float* __restrict__ gate,
    const float* __restrict__ Vx,
    float* __restrict__ out)
{
    __shared__ v4f sA[8 * 32];
    __shared__ v4f sG[8 * 32];

    const int bi = blockIdx.x;            // flattened (b,i), 0..2047
    const int b  = bi >> 8;               // bi / Nn
    const int g  = threadIdx.x >> 5;      // j-group 0..7
    const int ql = threadIdx.x & 31;      // channel-quad lane
    const int c  = ql * 4;                // channel base

    const float* grow = gate + (size_t)bi * Nn * Dd + c;
    const float* vrow = Vx   + (size_t)b  * Nn * Dd + c;

    v4f acc = {};
    v4f gs  = {};
    #pragma unroll 4
    for (int j = g; j < Nn; j += 8) {
        v4f gv = __builtin_nontemporal_load((const v4f*)(grow + (size_t)j * Dd));
        v4f vv = *(const v4f*)(vrow + (size_t)j * Dd);
        acc += gv * vv;
        gs  += gv;
    }
    sA[g * 32 + ql] = acc;
    sG[g * 32 + ql] = gs;
    __syncthreads();

    if (threadIdx.x < Dd) {
        const int ch = threadIdx.x;
        const float* fA = (const float*)sA;
        const float* fG = (const float*)sG;
        float a = 0.f, s = 0.f;
        #pragma unroll
        for (int gg = 0; gg < 8; ++gg) {
            a += fA[gg * Dd + ch];
            s += fG[gg * Dd + ch];
        }
        const size_t o = (size_t)bi * Dd + ch;
        out[o] = out[o] + a / (1e-20f + s);   // out already holds Ux
    }
}

// ---------------------------------------------------------------------------
extern "C" void kernel_launch(void* const* d_in, const int* in_sizes, int n_in,
                              void* d_out, int out_size, void* d_ws, size_t ws_size,
                              hipStream_t stream) {
    const float* x    = (const float*)d_in[0];
    const float* gate = (const float*)d_in[1];
    const float* Wu   = (const float*)d_in[2];
    const float* bu   = (const float*)d_in[3];
    const float* Wv   = (const float*)d_in[4];
    const float* bv   = (const float*)d_in[5];
    float* out = (float*)d_out;
    float* Vx  = (float*)d_ws;            // B*N*D floats = 1 MB scratch

    // 1) WMMA projections: 1024 tiles, 4 waves/block -> 256 blocks.
    proj_wmma_kernel<<<(MT * NT) / 4, 128, 0, stream>>>(x, Wu, bu, Wv, bv, out, Vx);

    // 2) Fused gated aggregation over the 268 MB edge_gate stream.
    gate_agg_kernel<<<R, 256, 0, stream>>>(gate, Vx, out);
}